// GCN_52767968199326
// MI455X (gfx1250) — compile-verified
//
#include <hip/hip_runtime.h>
#include <hip/hip_bf16.h>

typedef __attribute__((ext_vector_type(2))) float v2f;
typedef __attribute__((ext_vector_type(8))) float v8f;

// ---------------------------------------------------------------------------
// Pass 1: degree with self-loop weight 1.0 folded into the init.
// ---------------------------------------------------------------------------
__global__ void k_init_deg(float* __restrict__ deg, int n) {
    int i = blockIdx.x * blockDim.x + threadIdx.x;
    if (i < n) deg[i] = 1.0f;                    // self-loop contributes weight 1
}

__global__ void k_deg_edges(const int* __restrict__ col, const float* __restrict__ ew,
                            float* __restrict__ deg, int e) {
    int i = blockIdx.x * blockDim.x + threadIdx.x;
    if (i < e) atomicAdd(&deg[col[i]], ew[i]);   // global_atomic_add_f32, L2-resident
}

__global__ void k_dis(float* __restrict__ deg_dis, int n) {
    int i = blockIdx.x * blockDim.x + threadIdx.x;
    if (i < n) {
        float d = deg_dis[i];
        deg_dis[i] = (d > 0.f) ? rsqrtf(d) : 0.f;   // in-place deg -> deg^-1/2
    }
}

// ---------------------------------------------------------------------------
// h1[n,16] = x[n,3] @ W1[3,16] via V_WMMA_F32_16X16X4_F32 (K padded 3->4).
// One wave per 16-node tile. Branch-free: all loads use clamped in-bounds
// addresses + value selects (v_cndmask); stores take a wave-uniform full-tile
// fast path (8 x global_store_b32 at immediate offsets from one base).
// A 16x4: lanes 0-15 -> K=0,1 ; lanes 16-31 -> K=2,3.
// ---------------------------------------------------------------------------
__global__ void k_gemm1(const float* __restrict__ x, const float* __restrict__ W1,
                        float* __restrict__ h1, int n, int ntiles) {
    int gtid = blockIdx.x * blockDim.x + threadIdx.x;
    int tile = gtid >> 5;
    int lane = threadIdx.x & 31;
    if (tile >= ntiles) return;                  // wave-uniform; EXEC full at WMMA
    int half = lane >> 4;                        // 0: K=0,1   1: K=2,3
    int m    = lane & 15;
    int row  = tile * 16 + m;
    int rowc = row < n ? row : n - 1;            // clamped, always in-bounds
    bool rv  = row < n;
    int k0   = half * 2;                         // 0 or 2 (both < 3: safe)

    float ax = x[(size_t)rowc * 3 + k0];         // K=0 or K=2
    float ay = x[(size_t)rowc * 3 + 1];          // K=1 (only valid for half==0)
    v2f a;
    a.x = rv ? ax : 0.f;
    a.y = (rv && half == 0) ? ay : 0.f;          // half==1 slot is K=3 pad -> 0

    float bx = W1[k0 * 16 + m];                  // rows 0 and 2 of W1
    float by = W1[16 + m];                       // row 1 of W1
    v2f b;
    b.x = bx;
    b.y = (half == 0) ? by : 0.f;                // K=3 pad row -> 0

    v8f c = {};
    c = __builtin_amdgcn_wmma_f32_16x16x4_f32(false, a, false, b, (short)0, c, false, false);

    // D: VGPR j holds row j (lanes 0-15) / row j+8 (lanes 16-31), col = m
    float* p = h1 + ((size_t)(tile * 16 + half * 8)) * 16 + m;
    if (tile * 16 + 16 <= n) {                   // wave-uniform full-tile fast path
#pragma unroll
        for (int j = 0; j < 8; ++j) p[(size_t)j * 16] = c[j];
    } else {
#pragma unroll
        for (int j = 0; j < 8; ++j)
            if (tile * 16 + half * 8 + j < n) p[(size_t)j * 16] = c[j];
    }
}

// acc1 init = self-loop message: h1[i] * dis[i]^2  (norm_self = dis*1*dis)
__global__ void k_self1(const float* __restrict__ h1, const float* __restrict__ dis,
                        float* __restrict__ acc, int n16) {
    int i = blockIdx.x * blockDim.x + threadIdx.x;
    if (i < n16) {
        float s = dis[i >> 4];
        acc[i] = h1[i] * s * s;
    }
}

// Edge scatter, layer 1: acc[col] += h1[row] * (dis[row]*ew*dis[col])
__global__ void k_edges1(const int* __restrict__ row, const int* __restrict__ col,
                         const float* __restrict__ ew, const float* __restrict__ dis,
                         const float* __restrict__ h1, float* __restrict__ acc, int e) {
    int i = blockIdx.x * blockDim.x + threadIdx.x;
    if (i >= e) return;
    int r = row[i], c = col[i];
    float nrm = dis[r] * ew[i] * dis[c];
    const float4* hr = (const float4*)(h1 + (size_t)r * 16);   // 4x global_load_b128
    float4 v0 = hr[0], v1 = hr[1], v2 = hr[2], v3 = hr[3];
    float* dst = acc + (size_t)c * 16;
    atomicAdd(dst +  0, v0.x * nrm); atomicAdd(dst +  1, v0.y * nrm);
    atomicAdd(dst +  2, v0.z * nrm); atomicAdd(dst +  3, v0.w * nrm);
    atomicAdd(dst +  4, v1.x * nrm); atomicAdd(dst +  5, v1.y * nrm);
    atomicAdd(dst +  6, v1.z * nrm); atomicAdd(dst +  7, v1.w * nrm);
    atomicAdd(dst +  8, v2.x * nrm); atomicAdd(dst +  9, v2.y * nrm);
    atomicAdd(dst + 10, v2.z * nrm); atomicAdd(dst + 11, v2.w * nrm);
    atomicAdd(dst + 12, v3.x * nrm); atomicAdd(dst + 13, v3.y * nrm);
    atomicAdd(dst + 14, v3.z * nrm); atomicAdd(dst + 15, v3.w * nrm);
}

__global__ void k_relu(float* __restrict__ acc, const float* __restrict__ b1, int n16) {
    int i = blockIdx.x * blockDim.x + threadIdx.x;
    if (i < n16) acc[i] = fmaxf(acc[i] + b1[i & 15], 0.f);
}

// ---------------------------------------------------------------------------
// h2[n,16 (7 used, rest exact 0)] = z[n,16] @ W2[16, 7 padded to 16 cols]
// Chain of 4 fp32 WMMAs (K = 16 = 4x4). Branch-free loads and stores.
// ---------------------------------------------------------------------------
__global__ void k_gemm2(const float* __restrict__ z, const float* __restrict__ W2,
                        float* __restrict__ h2, int n, int ntiles) {
    int gtid = blockIdx.x * blockDim.x + threadIdx.x;
    int tile = gtid >> 5;
    int lane = threadIdx.x & 31;
    if (tile >= ntiles) return;
    int half = lane >> 4;
    int m    = lane & 15;
    int row  = tile * 16 + m;
    int rowc = row < n ? row : n - 1;
    bool rv  = row < n;
    int mc   = m < 7 ? m : 6;                    // clamped W2 column
    bool mv  = m < 7;

    v8f c = {};
#pragma unroll
    for (int s = 0; s < 4; ++s) {
        int k = s * 4 + half * 2;
        float ax = z[(size_t)rowc * 16 + k + 0];
        float ay = z[(size_t)rowc * 16 + k + 1];
        v2f a;
        a.x = rv ? ax : 0.f;
        a.y = rv ? ay : 0.f;
        float bx = W2[(k + 0) * 7 + mc];
        float by = W2[(k + 1) * 7 + mc];
        v2f b;
        b.x = mv ? bx : 0.f;                     // zero-pad cols 7..15
        b.y = mv ? by : 0.f;
        c = __builtin_amdgcn_wmma_f32_16x16x4_f32(false, a, false, b, (short)0, c, false, false);
    }

    float* p = h2 + ((size_t)(tile * 16 + half * 8)) * 16 + m;
    if (tile * 16 + 16 <= n) {                   // wave-uniform fast path
#pragma unroll
        for (int j = 0; j < 8; ++j) p[(size_t)j * 16] = c[j];   // cols 7..15 store 0
    } else {
#pragma unroll
        for (int j = 0; j < 8; ++j)
            if (tile * 16 + half * 8 + j < n) p[(size_t)j * 16] = c[j];
    }
}

// out init = bias + self-loop message
__global__ void k_out_init(const float* __restrict__ h2, const float* __restrict__ dis,
                           const float* __restrict__ b2, float* __restrict__ out, int n7) {
    int i = blockIdx.x * blockDim.x + threadIdx.x;
    if (i < n7) {
        int node = i / 7, f = i % 7;
        float s = dis[node];
        out[i] = b2[f] + h2[(size_t)node * 16 + f] * s * s;
    }
}

// Edge scatter, layer 2: out[col] += h2[row] * norm (7 features)
__global__ void k_edges2(const int* __restrict__ row, const int* __restrict__ col,
                         const float* __restrict__ ew, const float* __restrict__ dis,
                         const float* __restrict__ h2, float* __restrict__ out, int e) {
    int i = blockIdx.x * blockDim.x + threadIdx.x;
    if (i >= e) return;
    int r = row[i], c = col[i];
    float nrm = dis[r] * ew[i] * dis[c];
    const float4* hr = (const float4*)(h2 + (size_t)r * 16);
    float4 v0 = hr[0], v1 = hr[1];               // h2[..][7] == 0, in-bounds
    float* dst = out + (size_t)c * 7;
    atomicAdd(dst + 0, v0.x * nrm); atomicAdd(dst + 1, v0.y * nrm);
    atomicAdd(dst + 2, v0.z * nrm); atomicAdd(dst + 3, v0.w * nrm);
    atomicAdd(dst + 4, v1.x * nrm); atomicAdd(dst + 5, v1.y * nrm);
    atomicAdd(dst + 6, v1.z * nrm);
}

extern "C" void kernel_launch(void* const* d_in, const int* in_sizes, int n_in,
                              void* d_out, int out_size, void* d_ws, size_t ws_size,
                              hipStream_t stream) {
    (void)n_in; (void)out_size; (void)ws_size;
    const float* x  = (const float*)d_in[0];
    const int*   ei = (const int*)d_in[1];
    const float* ew = (const float*)d_in[2];
    const float* W1 = (const float*)d_in[3];
    const float* b1 = (const float*)d_in[4];
    const float* W2 = (const float*)d_in[5];
    const float* b2 = (const float*)d_in[6];
    float* out = (float*)d_out;

    int n = in_sizes[0] / 3;          // 200000
    int e = in_sizes[2];              // 5000000
    const int* row = ei;              // edge_index[0]
    const int* col = ei + e;          // edge_index[1]

    // workspace: dis[n] | h1[16n] | acc1[16n] | h2[16n]  (~39 MB, L2-resident)
    float* ws   = (float*)d_ws;
    float* dis  = ws;
    float* h1   = dis  + (size_t)n;
    float* acc1 = h1   + (size_t)n * 16;
    float* h2   = acc1 + (size_t)n * 16;

    const int B = 256;
    int ntiles  = (n + 15) / 16;
    int gemmT   = ntiles * 32;

    k_init_deg <<<(n + B - 1) / B, B, 0, stream>>>(dis, n);
    k_deg_edges<<<(e + B - 1) / B, B, 0, stream>>>(col, ew, dis, e);
    k_dis      <<<(n + B - 1) / B, B, 0, stream>>>(dis, n);

    k_gemm1    <<<(gemmT + B - 1) / B, B, 0, stream>>>(x, W1, h1, n, ntiles);
    k_self1    <<<(n * 16 + B - 1) / B, B, 0, stream>>>(h1, dis, acc1, n * 16);
    k_edges1   <<<(e + B - 1) / B, B, 0, stream>>>(row, col, ew, dis, h1, acc1, e);
    k_relu     <<<(n * 16 + B - 1) / B, B, 0, stream>>>(acc1, b1, n * 16);

    k_gemm2    <<<(gemmT + B - 1) / B, B, 0, stream>>>(acc1, W2, h2, n, ntiles);
    k_out_init <<<(n * 7 + B - 1) / B, B, 0, stream>>>(h2, dis, b2, out, n * 7);
    k_edges2   <<<(e + B - 1) / B, B, 0, stream>>>(row, col, ew, dis, h2, out, e);
}